// SimpleSSM_87205015978000
// MI455X (gfx1250) — compile-verified
//
#include <hip/hip_runtime.h>
#include <hip/hip_bf16.h>
#include <stdint.h>

typedef __attribute__((ext_vector_type(2))) float v2f;
typedef __attribute__((ext_vector_type(8))) float v8f;

#define BATCH   4
#define LSEQ    4096
#define DM      1024
#define NS      16
#define TBLK    16
#define CHUNK   256
#define NCH     (LSEQ / CHUNK)      /* 16 */
#define DGRP    16
#define PITCH   17

/* workspace layout (float offsets), total ~11.1 MiB */
#define OFF_M    0
#define OFF_W    (OFF_M + DM*256)
#define OFF_V    (OFF_W + DM*256)
#define OFF_A16  (OFF_V + DM*256)
#define OFF_AC   (OFF_A16 + DM*16)
#define OFF_QLOC (OFF_AC + DM*16)
#define OFF_QIN  (OFF_QLOC + BATCH*DM*NCH*NS)

// ---------------------------------------------------------------------------
// Precompute per-channel matrices:
//   M[t][s] = sum_n cb_n a_n^(t-s) (t>=s)   (block-local Toeplitz kernel)
//   W[t][n] = cb_n a_n^(t+1)                (carry-in -> output)
//   V[n][s] = a_n^(15-s)                    (input -> block-end state)
//   a16 = a^16, aC = a^256
// ---------------------------------------------------------------------------
__global__ void ssm_precompute(const float* __restrict__ Ag, const float* __restrict__ Bg,
                               const float* __restrict__ Cg, const float* __restrict__ dg,
                               float* __restrict__ ws) {
    int d = blockIdx.x * blockDim.x + threadIdx.x;
    if (d >= DM) return;
    float dt = log1pf(__expf(dg[d]));   /* softplus */
    float a[NS], cb[NS];
#pragma unroll
    for (int n = 0; n < NS; ++n) {
        a[n]  = __expf(dt * Ag[d*NS + n]);
        cb[n] = dt * Bg[d*NS + n] * Cg[d*NS + n];
    }
    float K[TBLK];
    {
        float pw[NS];
#pragma unroll
        for (int n = 0; n < NS; ++n) pw[n] = cb[n];
#pragma unroll
        for (int t = 0; t < TBLK; ++t) {
            float s = 0.f;
#pragma unroll
            for (int n = 0; n < NS; ++n) { s += pw[n]; pw[n] *= a[n]; }
            K[t] = s;
        }
    }
    float* M = ws + OFF_M + (size_t)d * 256;
#pragma unroll
    for (int t = 0; t < 16; ++t)
#pragma unroll
        for (int s = 0; s < 16; ++s)
            M[t*16 + s] = (t >= s) ? K[t - s] : 0.f;
    {
        float pw[NS];
#pragma unroll
        for (int n = 0; n < NS; ++n) pw[n] = a[n];
        float* W = ws + OFF_W + (size_t)d * 256;
#pragma unroll
        for (int t = 0; t < 16; ++t) {
#pragma unroll
            for (int n = 0; n < NS; ++n) { W[t*16 + n] = cb[n] * pw[n]; pw[n] *= a[n]; }
        }
    }
    {
        float* V = ws + OFF_V + (size_t)d * 256;
#pragma unroll
        for (int n = 0; n < NS; ++n) {
            float p = 1.f;
#pragma unroll
            for (int s = 15; s >= 0; --s) { V[n*16 + s] = p; p *= a[n]; }
        }
    }
#pragma unroll
    for (int n = 0; n < NS; ++n) {
        float a2 = a[n]*a[n], a4 = a2*a2, a8 = a4*a4, a16 = a8*a8;
        float a32 = a16*a16, a64 = a32*a32, a128 = a64*a64, a256 = a128*a128;
        ws[OFF_A16 + d*16 + n] = a16;
        ws[OFF_AC  + d*16 + n] = a256;
    }
}

// ---------------------------------------------------------------------------
// Coalesced async stage of a 256(t) x 16(d) f32 tile into LDS (pitch-padded).
// Uses CDNA5 async-to-LDS path (ASYNCcnt).
// ---------------------------------------------------------------------------
__device__ __forceinline__ void stage_x_async(const float* __restrict__ xg, float* sx) {
#pragma unroll
    for (int r = 0; r < 8; ++r) {
        int e  = threadIdx.x + 512 * r;    /* 0..4095 */
        int tl = e >> 4;
        int dl = e & 15;
        const float* gp = xg + (size_t)tl * DM + dl;
        unsigned loff = (unsigned)(uintptr_t)(sx + tl * PITCH + dl);
        asm volatile("global_load_async_to_lds_b32 %0, %1, off"
                     :: "v"(loff), "v"(gp) : "memory");
    }
    asm volatile("s_wait_asynccnt 0x0" ::: "memory");
    __syncthreads();
}

// ---------------------------------------------------------------------------
// Pass 1: per (b, d, chunk) wave -> chunk-local end state (sigma from 0 init)
//   U = V @ X (WMMA), then qloc[n] = sum_blk a16^(15-blk) * U[n,blk]
// ---------------------------------------------------------------------------
__global__ __launch_bounds__(512) void ssm_pass1(const float* __restrict__ x,
                                                 float* __restrict__ ws) {
    __shared__ float sx[CHUNK * PITCH];
    int bid   = blockIdx.x;
    int chunk = bid & (NCH - 1);
    int dg    = (bid >> 4) & 63;
    int b     = bid >> 10;
    const float* xg = x + ((size_t)b * LSEQ + (size_t)chunk * CHUNK) * DM + dg * DGRP;
    stage_x_async(xg, sx);

    int w = threadIdx.x >> 5, lid = threadIdx.x & 31;
    int col = lid & 15, hi = lid >> 4, row = col;
    int d = dg * DGRP + w;

    const float* Vd = ws + OFF_V + (size_t)d * 256;
    v8f u = {};
#pragma unroll
    for (int kk = 0; kk < 4; ++kk) {
        int kb = 4*kk + 2*hi;
        v2f av; av.x = Vd[row*16 + kb]; av.y = Vd[row*16 + kb + 1];
        v2f bx; bx.x = sx[(kb     + 16*col) * PITCH + w];
                bx.y = sx[(kb + 1 + 16*col) * PITCH + w];
        u = __builtin_amdgcn_wmma_f32_16x16x4_f32(false, av, false, bx,
                                                  (short)0, u, false, false);
    }
    const float* a16p = ws + OFF_A16 + (size_t)d * 16;
    float f1[8], f2[8], f4[8], f8[8];
#pragma unroll
    for (int r = 0; r < 8; ++r) {
        float v = a16p[r + 8*hi];
        f1[r] = v; f2[r] = v*v; f4[r] = f2[r]*f2[r]; f8[r] = f4[r]*f4[r];
    }
    int m = 15 - col;
    float* qloc = ws + OFF_QLOC + (((size_t)b * DM + d) * NCH + chunk) * NS;
#pragma unroll
    for (int r = 0; r < 8; ++r) {
        float fb = 1.f;
        fb *= (m & 1) ? f1[r] : 1.f;
        fb *= (m & 2) ? f2[r] : 1.f;
        fb *= (m & 4) ? f4[r] : 1.f;
        fb *= (m & 8) ? f8[r] : 1.f;
        float s = u[r] * fb;
        s += __shfl_xor(s, 1, 16);
        s += __shfl_xor(s, 2, 16);
        s += __shfl_xor(s, 4, 16);
        s += __shfl_xor(s, 8, 16);
        if (col == 0) qloc[r + 8*hi] = s;
    }
}

// ---------------------------------------------------------------------------
// Pass 2: tiny inter-chunk scan per (b,d):  qin[c] = state; state = aC*state + qloc[c]
// ---------------------------------------------------------------------------
__global__ void ssm_pass2(float* __restrict__ ws) {
    int idx = blockIdx.x * blockDim.x + threadIdx.x;   /* b*DM + d */
    if (idx >= BATCH * DM) return;
    int d = idx & (DM - 1);
    const float* aC = ws + OFF_AC + (size_t)d * 16;
    float q[NS];
#pragma unroll
    for (int n = 0; n < NS; ++n) q[n] = 0.f;
    size_t base = (size_t)idx * NCH * NS;
    float* qloc = ws + OFF_QLOC + base;
    float* qin  = ws + OFF_QIN  + base;
    for (int c = 0; c < NCH; ++c) {
#pragma unroll
        for (int n = 0; n < NS; ++n) {
            qin[c*NS + n] = q[n];
            q[n] = fmaf(aC[n], q[n], qloc[c*NS + n]);
        }
    }
}

// ---------------------------------------------------------------------------
// Pass 3: per (b, d, chunk) wave -> outputs
//   U = V@X ; sigma_in per block via decayed lane-scan of U (+ chunk carry)
//   Y = M@X + W@sigma_in  (all WMMA), store via LDS transpose (coalesced)
// ---------------------------------------------------------------------------
__global__ __launch_bounds__(512) void ssm_pass3(const float* __restrict__ x,
                                                 float* __restrict__ y,
                                                 float* __restrict__ ws) {
    __shared__ float sx[CHUNK * PITCH];
    __shared__ float sy[CHUNK * PITCH];
    __shared__ float sq[16 * 256];          /* per-wave 16x16 sigma_in tile */
    int bid   = blockIdx.x;
    int chunk = bid & (NCH - 1);
    int dg    = (bid >> 4) & 63;
    int b     = bid >> 10;
    const float* xg = x + ((size_t)b * LSEQ + (size_t)chunk * CHUNK) * DM + dg * DGRP;
    stage_x_async(xg, sx);

    int w = threadIdx.x >> 5, lid = threadIdx.x & 31;
    int col = lid & 15, hi = lid >> 4, row = col;
    int d = dg * DGRP + w;

    const float* Vd = ws + OFF_V + (size_t)d * 256;
    const float* Md = ws + OFF_M + (size_t)d * 256;
    const float* Wd = ws + OFF_W + (size_t)d * 256;

    v8f u = {}, acc = {};
    v2f bx[4];
#pragma unroll
    for (int kk = 0; kk < 4; ++kk) {
        int kb = 4*kk + 2*hi;
        bx[kk].x = sx[(kb     + 16*col) * PITCH + w];
        bx[kk].y = sx[(kb + 1 + 16*col) * PITCH + w];
    }
#pragma unroll
    for (int kk = 0; kk < 4; ++kk) {
        int kb = 4*kk + 2*hi;
        v2f av; av.x = Vd[row*16 + kb]; av.y = Vd[row*16 + kb + 1];
        u = __builtin_amdgcn_wmma_f32_16x16x4_f32(false, av, false, bx[kk],
                                                  (short)0, u, false, false);
        v2f am; am.x = Md[row*16 + kb]; am.y = Md[row*16 + kb + 1];
        acc = __builtin_amdgcn_wmma_f32_16x16x4_f32(false, am, false, bx[kk],
                                                    (short)0, acc, false, false);
    }
    /* per-row a16 powers for the decayed lane scan */
    const float* a16p = ws + OFF_A16 + (size_t)d * 16;
    float f1[8], f2[8], f4[8], f8[8];
#pragma unroll
    for (int r = 0; r < 8; ++r) {
        float v = a16p[r + 8*hi];
        f1[r] = v; f2[r] = v*v; f4[r] = f2[r]*f2[r]; f8[r] = f4[r]*f4[r];
    }
    const float* qin = ws + OFF_QIN + (((size_t)b * DM + d) * NCH + chunk) * NS;
    float* sqw = sq + w * 256;
#pragma unroll
    for (int r = 0; r < 8; ++r) {
        float v = u[r], t;
        t = __shfl_up(v, 1, 16);  v = fmaf(f1[r], (col >= 1) ? t : 0.f, v);
        t = __shfl_up(v, 2, 16);  v = fmaf(f2[r], (col >= 2) ? t : 0.f, v);
        t = __shfl_up(v, 4, 16);  v = fmaf(f4[r], (col >= 4) ? t : 0.f, v);
        t = __shfl_up(v, 8, 16);  v = fmaf(f8[r], (col >= 8) ? t : 0.f, v);
        t = __shfl_up(v, 1, 16);
        float ex  = (col >= 1) ? t : 0.f;        /* exclusive decayed prefix */
        float fbc = 1.f;                          /* a16^col for chunk carry  */
        fbc *= (col & 1) ? f1[r] : 1.f;
        fbc *= (col & 2) ? f2[r] : 1.f;
        fbc *= (col & 4) ? f4[r] : 1.f;
        fbc *= (col & 8) ? f8[r] : 1.f;
        float qr = fmaf(fbc, qin[r + 8*hi], ex);
        sqw[(r + 8*hi) * 16 + col] = qr;          /* C-layout -> LDS */
    }
    asm volatile("s_wait_dscnt 0x0" ::: "memory"); /* wave-private LDS transpose */
#pragma unroll
    for (int kk = 0; kk < 4; ++kk) {
        int kb = 4*kk + 2*hi;
        v2f bq; bq.x = sqw[kb * 16 + col]; bq.y = sqw[(kb + 1) * 16 + col];
        v2f aw; aw.x = Wd[row*16 + kb];    aw.y = Wd[row*16 + kb + 1];
        acc = __builtin_amdgcn_wmma_f32_16x16x4_f32(false, aw, false, bq,
                                                    (short)0, acc, false, false);
    }
    /* stage Y through LDS for coalesced global stores */
#pragma unroll
    for (int r = 0; r < 8; ++r)
        sy[((r + 8*hi) + 16*col) * PITCH + w] = acc[r];
    __syncthreads();
    float* yg = y + ((size_t)b * LSEQ + (size_t)chunk * CHUNK) * DM + dg * DGRP;
#pragma unroll
    for (int r8 = 0; r8 < 8; ++r8) {
        int e = threadIdx.x + 512 * r8;
        int tl = e >> 4, dl = e & 15;
        yg[(size_t)tl * DM + dl] = sy[tl * PITCH + dl];
    }
}

extern "C" void kernel_launch(void* const* d_in, const int* in_sizes, int n_in,
                              void* d_out, int out_size, void* d_ws, size_t ws_size,
                              hipStream_t stream) {
    (void)in_sizes; (void)n_in; (void)out_size; (void)ws_size;
    const float* x  = (const float*)d_in[0];
    const float* A  = (const float*)d_in[1];
    const float* Bm = (const float*)d_in[2];
    const float* Cm = (const float*)d_in[3];
    const float* dl = (const float*)d_in[4];
    float* y  = (float*)d_out;
    float* ws = (float*)d_ws;

    ssm_precompute<<<DM / 256, 256, 0, stream>>>(A, Bm, Cm, dl, ws);
    ssm_pass1<<<BATCH * (DM / DGRP) * NCH, 512, 0, stream>>>(x, ws);
    ssm_pass2<<<(BATCH * DM) / 256, 256, 0, stream>>>(ws);
    ssm_pass3<<<BATCH * (DM / DGRP) * NCH, 512, 0, stream>>>(x, y, ws);
}